// Deep_Mem_AbsLocs_40089224741406
// MI455X (gfx1250) — compile-verified
//
#include <hip/hip_runtime.h>

// -----------------------------------------------------------------------------
// Deep_Mem_AbsLocs: 6-D scatter-add histogram.
//   code(p) = ((clip(rint(x+32),0,64)*65 + clip(rint(y+32),0,64))*2
//              + clip(rint(tex),0,1))                  in [0, 8450)
//   for each edge (a,b):  out[code(a)*8450 + code(b)] += 1
//                         out[code(b)*8450 + code(a)] += 1   (reversed concat)
// Output: 65*65*2*65*65*2 = 71,402,500 f32 (285.6 MB). Memory/atomic bound.
// No matmul content -> no WMMA; optimize the HBM/L2-atomic path instead.
// -----------------------------------------------------------------------------

typedef __attribute__((ext_vector_type(4))) float     v4f;
typedef __attribute__((ext_vector_type(2))) float     v2f;
typedef __attribute__((ext_vector_type(2))) int       v2i;

#define GRID_N   65
#define NBINS    (GRID_N * GRID_N * 2)   // 8450 codes per endpoint
#define BLOCK    256                     // 8 wave32 waves

// ---------------------------------------------------------------------------
// Kernel 1: zero the accumulator with b128 stores.
// Default (RT) temporal hint on purpose: the freshly zeroed lines stay dirty
// in the 192MB L2, so the following atomic phase RMWs mostly hit in L2.
// ---------------------------------------------------------------------------
__global__ __launch_bounds__(BLOCK)
void zero_out_kernel(v4f* __restrict__ out, long n4) {
    long i = (long)blockIdx.x * BLOCK + threadIdx.x;
    if (i < n4) {
        v4f z = {0.0f, 0.0f, 0.0f, 0.0f};
        out[i] = z;                       // global_store_b128
    }
}

// ---------------------------------------------------------------------------
// Per-point code. rintf under default round-to-nearest-even == jnp.round
// (banker's rounding). Inputs are bounded (|x|<=32, tex in [0,1)) so the
// int conversion cannot overflow; clamp after convert.
// ---------------------------------------------------------------------------
__device__ __forceinline__ int point_code(const v2f* __restrict__ pts,
                                          const float* __restrict__ tex,
                                          int idx) {
    v2f   p = pts[idx];       // 8B, L2-resident after warmup (pts = 1.6MB)
    float t = tex[idx];       // 4B, L2-resident              (tex = 0.8MB)
    int ix = (int)rintf(p.x + 32.0f);
    int iy = (int)rintf(p.y + 32.0f);
    int it = (int)rintf(t);
    ix = ix < 0 ? 0 : (ix > GRID_N - 1 ? GRID_N - 1 : ix);
    iy = iy < 0 ? 0 : (iy > GRID_N - 1 ? GRID_N - 1 : iy);
    it = it < 0 ? 0 : (it > 1 ? 1 : it);
    return (ix * GRID_N + iy) * 2 + it;
}

// ---------------------------------------------------------------------------
// Kernel 2: one thread per original edge; two non-returning FP32 atomics
// (forward bin + reversed bin). Edge stream is read exactly once ->
// non-temporal load keeps L2 free for the output bins + point tables.
// ---------------------------------------------------------------------------
__global__ __launch_bounds__(BLOCK)
void edge_scatter_kernel(const v2i* __restrict__ edges,   // (n_edges, 2) int32
                         const v2f* __restrict__ pts,     // (N_PTS, 2) f32
                         const float* __restrict__ tex,   // (N_PTS,)  f32
                         float* __restrict__ out,         // 8450*8450 f32
                         long n_edges) {
    long i = (long)blockIdx.x * BLOCK + threadIdx.x;
    if (i >= n_edges) return;

    v2i e = __builtin_nontemporal_load(&edges[i]);  // global_load_b64 th:NT
    int a = e.x;
    int b = e.y;

    int ca = point_code(pts, tex, a);
    int cb = point_code(pts, tex, b);

    size_t bin_fwd = (size_t)ca * NBINS + (size_t)cb;
    size_t bin_rev = (size_t)cb * NBINS + (size_t)ca;

    // Non-returning, relaxed, device-scope FP32 atomic ->
    // global_atomic_add_f32 (no VDST), tracked on STOREcnt.
    __hip_atomic_fetch_add(out + bin_fwd, 1.0f,
                           __ATOMIC_RELAXED, __HIP_MEMORY_SCOPE_AGENT);
    __hip_atomic_fetch_add(out + bin_rev, 1.0f,
                           __ATOMIC_RELAXED, __HIP_MEMORY_SCOPE_AGENT);
}

// ---------------------------------------------------------------------------
// Launch: zero (285.6MB, ~12us @ 23.3TB/s) then scatter (~19us HBM-side).
// Same-stream ordering serializes the two phases. Deterministic: every call
// re-zeros d_out and re-accumulates identical work.
// ---------------------------------------------------------------------------
extern "C" void kernel_launch(void* const* d_in, const int* in_sizes, int n_in,
                              void* d_out, int out_size, void* d_ws, size_t ws_size,
                              hipStream_t stream) {
    const v2f*   pts   = (const v2f*)d_in[0];   // in_sizes[0] = 400000 floats
    const float* tex   = (const float*)d_in[1]; // in_sizes[1] = 200000 floats
    const v2i*   edges = (const v2i*)d_in[2];   // in_sizes[2] = 3,200,000 ints
    // d_in[3] is the zero-initialized 'mem' input; we rebuild it in d_out.

    float* out = (float*)d_out;

    long n_edges = (long)in_sizes[2] / 2;       // 1,600,000
    long n4      = (long)out_size / 4;          // 71,402,500 / 4 = 17,850,625 (exact)

    long zero_blocks = (n4 + BLOCK - 1) / BLOCK;
    zero_out_kernel<<<(unsigned)zero_blocks, BLOCK, 0, stream>>>((v4f*)out, n4);

    long edge_blocks = (n_edges + BLOCK - 1) / BLOCK;
    edge_scatter_kernel<<<(unsigned)edge_blocks, BLOCK, 0, stream>>>(
        edges, pts, tex, out, n_edges);
}